// NeuralHMM_8186207666596
// MI455X (gfx1250) — compile-verified
//
#include <hip/hip_runtime.h>

#define TLEN 2048
#define H 256
#define V 1024

typedef float v2f __attribute__((ext_vector_type(2)));
typedef float v8f __attribute__((ext_vector_type(8)));

// ---------------- block reduction helpers (blockDim.x == 256) ----------------
__device__ __forceinline__ float block_reduce_sum(float v, float* red) {
    int tid = threadIdx.x;
    __syncthreads();              // protect any prior use of red
    red[tid] = v;
    __syncthreads();
    for (int off = 128; off > 0; off >>= 1) {
        if (tid < off) red[tid] += red[tid + off];
        __syncthreads();
    }
    float r = red[0];
    __syncthreads();
    return r;
}

__device__ __forceinline__ float block_reduce_max(float v, float* red) {
    int tid = threadIdx.x;
    __syncthreads();
    red[tid] = v;
    __syncthreads();
    for (int off = 128; off > 0; off >>= 1) {
        if (tid < off) red[tid] = fmaxf(red[tid], red[tid + off]);
        __syncthreads();
    }
    float r = red[0];
    __syncthreads();
    return r;
}

// ---------------- softmax prep kernels ----------------
__global__ __launch_bounds__(256) void prior_softmax(const float* __restrict__ sp,
                                                     float* __restrict__ pi,
                                                     float* __restrict__ logpi) {
    __shared__ float red[256];
    int j = threadIdx.x;
    float x = sp[j];
    float m = block_reduce_max(x, red);
    float e = expf(x - m);
    float s = block_reduce_sum(e, red);
    float lp = x - m - logf(s);
    pi[j] = e / s;
    logpi[j] = lp;
}

__global__ __launch_bounds__(256) void trans_softmax(const float* __restrict__ u,
                                                     float* __restrict__ trans,
                                                     float* __restrict__ transT,
                                                     float* __restrict__ tlogt) {
    __shared__ float red[256];
    int i = blockIdx.x, j = threadIdx.x;
    float x = u[i * H + j];
    float m = block_reduce_max(x, red);
    float e = expf(x - m);
    float s = block_reduce_sum(e, red);
    float lp = x - m - logf(s);
    float p = e / s;
    trans[i * H + j] = p;
    transT[j * H + i] = p;
    tlogt[i * H + j] = p * lp;
}

__global__ __launch_bounds__(256) void emiss_softmax(const float* __restrict__ u,
                                                     float* __restrict__ emissP) {
    __shared__ float red[256];
    int i = blockIdx.x, j = threadIdx.x;
    float x0 = u[i * V + j];
    float x1 = u[i * V + j + 256];
    float x2 = u[i * V + j + 512];
    float x3 = u[i * V + j + 768];
    float ml = fmaxf(fmaxf(x0, x1), fmaxf(x2, x3));
    float m = block_reduce_max(ml, red);
    float e0 = expf(x0 - m), e1 = expf(x1 - m), e2 = expf(x2 - m), e3 = expf(x3 - m);
    float s = block_reduce_sum(e0 + e1 + e2 + e3, red);
    float inv = 1.f / s;
    emissP[i * V + j] = e0 * inv;
    emissP[i * V + j + 256] = e1 * inv;
    emissP[i * V + j + 512] = e2 * inv;
    emissP[i * V + j + 768] = e3 * inv;
}

// ---------------- WMMA GEMM 1: b = obs @ emissP^T, logb = log(b) ----------------
// C[t,n] = sum_v obs[t,v] * emissP[n,v]. Tiles 16x16, K-steps of 4 (f32 WMMA).
__global__ __launch_bounds__(256) void logb_gemm(const float* __restrict__ obs,
                                                 const float* __restrict__ emissP,
                                                 float* __restrict__ logb,
                                                 float* __restrict__ bprob) {
    int wave = threadIdx.x >> 5;
    int lane = threadIdx.x & 31;
    int tileIdx = blockIdx.x * 8 + wave;            // 2048 tiles: 128 (M) x 16 (N)
    int tileM = (tileIdx >> 4) << 4;                // T direction
    int tileN = (tileIdx & 15) << 4;                // H direction
    int l15 = lane & 15;
    int khalf = (lane >> 4) << 1;                   // lanes 0-15: K=0,1 ; lanes 16-31: K=2,3
    const float* aptr = obs + (size_t)(tileM + l15) * V + khalf;
    const float* bptr = emissP + (size_t)(tileN + l15) * V + khalf;
    v8f acc = {};
    for (int k = 0; k < V; k += 4) {
        v2f a = {aptr[k], aptr[k + 1]};
        v2f b = {bptr[k], bptr[k + 1]};
        acc = __builtin_amdgcn_wmma_f32_16x16x4_f32(false, a, false, b,
                                                    (short)0, acc, false, false);
    }
    int m0 = (lane >> 4) << 3;                      // C/D: lanes 16-31 hold rows M+8
    for (int r = 0; r < 8; ++r) {
        int row = tileM + m0 + r;
        int col = tileN + l15;
        float v = acc[r];
        bprob[row * H + col] = v;
        logb[row * H + col] = logf(v);
    }
}

// ---------------- forward scan (scaled alpha recursion, one workgroup) ----------------
// CDNA5 WGP has 320 KB LDS: the full 256 KB transition matrix is staged in LDS once,
// so each of the 2048 serial steps runs on LDS latency instead of L2 round-trips.
// Bank analysis: tmat[i*256 + j] with j = lane -> bank = j % 64, conflict-free.
__global__ __launch_bounds__(256) void forward_scan(const float* __restrict__ pi,
                                                    const float* __restrict__ trans,
                                                    const float* __restrict__ b,
                                                    float* __restrict__ alpha) {
    __shared__ float tmat[H * H];                   // 256 KB
    __shared__ float a_s[256];
    __shared__ float red[256];
    int j = threadIdx.x;
    {
        const float4* src = (const float4*)trans;
        float4* dst = (float4*)tmat;
        for (int idx = j; idx < (H * H) / 4; idx += 256) dst[idx] = src[idx];
    }
    float aj = pi[j] * b[j];
    float tot = block_reduce_sum(aj, red);          // also covers the LDS-fill barrier
    aj /= tot;
    a_s[j] = aj;
    alpha[j] = aj;
    __syncthreads();
    for (int t = 1; t < TLEN; ++t) {
        float s = 0.f;
#pragma unroll 8
        for (int i = 0; i < H; ++i) s = fmaf(a_s[i], tmat[i * H + j], s);
        s *= b[t * H + j];
        float tt = block_reduce_sum(s, red);
        s /= tt;
        a_s[j] = s;                                  // reduce barriers cover prior reads
        alpha[t * H + j] = s;
        __syncthreads();
    }
}

// ---------------- backward scan (scaled; also emits w[t] = b[t] * beta_hat[t]) ---------
__global__ __launch_bounds__(256) void backward_scan(const float* __restrict__ transT,
                                                     const float* __restrict__ b,
                                                     float* __restrict__ beta,
                                                     float* __restrict__ w) {
    __shared__ float tmat[H * H];                   // 256 KB (transT)
    __shared__ float tmp_s[256];
    __shared__ float red[256];
    int i = threadIdx.x;
    {
        const float4* src = (const float4*)transT;
        float4* dst = (float4*)tmat;
        for (int idx = i; idx < (H * H) / 4; idx += 256) dst[idx] = src[idx];
    }
    beta[(TLEN - 1) * H + i] = 1.f;
    w[(TLEN - 1) * H + i] = b[(TLEN - 1) * H + i];
    float bn = 1.f;                                  // beta_hat[t+1][i]
    __syncthreads();                                 // LDS fill complete
    for (int t = TLEN - 2; t >= 0; --t) {
        float bt = b[t * H + i];
        tmp_s[i] = bt * bn;                          // b[t][j] * beta_hat[t+1][j]
        __syncthreads();
        float s = 0.f;
#pragma unroll 8
        for (int j = 0; j < H; ++j) s = fmaf(tmp_s[j], tmat[j * H + i], s);
        float tt = block_reduce_sum(s, red);
        s /= tt;
        bn = s;
        beta[t * H + i] = s;
        w[t * H + i] = bt * s;
    }
}

// ---------------- WMMA GEMM 2: PT[s,i] = (T w_s)[i], QT[s,i] = ((T.logT) w_s)[i] -------
__global__ __launch_bounds__(256) void xi_gemm(const float* __restrict__ trans,
                                               const float* __restrict__ tlogt,
                                               const float* __restrict__ w,
                                               float* __restrict__ PT,
                                               float* __restrict__ QT) {
    int wave = threadIdx.x >> 5;
    int lane = threadIdx.x & 31;
    int tileIdx = blockIdx.x * 8 + wave;            // 2048 tiles: 16 (M=i) x 128 (N=s)
    int tileM = (tileIdx >> 7) << 4;                // i direction (H)
    int tileN = (tileIdx & 127) << 4;               // s direction (T)
    int l15 = lane & 15;
    int khalf = (lane >> 4) << 1;
    const float* aP = trans + (size_t)(tileM + l15) * H + khalf;
    const float* aQ = tlogt + (size_t)(tileM + l15) * H + khalf;
    const float* bp = w + (size_t)(tileN + l15) * H + khalf;   // B[k][s] = w[s*H + k]
    v8f accP = {};
    v8f accQ = {};
    for (int k = 0; k < H; k += 4) {
        v2f bfrag = {bp[k], bp[k + 1]};
        v2f a1 = {aP[k], aP[k + 1]};
        v2f a2 = {aQ[k], aQ[k + 1]};
        accP = __builtin_amdgcn_wmma_f32_16x16x4_f32(false, a1, false, bfrag,
                                                     (short)0, accP, false, false);
        accQ = __builtin_amdgcn_wmma_f32_16x16x4_f32(false, a2, false, bfrag,
                                                     (short)0, accQ, false, false);
    }
    int m0 = (lane >> 4) << 3;
    for (int r = 0; r < 8; ++r) {
        int i = tileM + m0 + r;                      // row index (i)
        int s = tileN + l15;                         // column index (s)
        PT[s * H + i] = accP[r];
        QT[s * H + i] = accQ[r];
    }
}

// ---------------- per-timestep reductions: gamma / emission / xi ratio -----------------
__global__ __launch_bounds__(256) void per_t_reduce(const float* __restrict__ alpha,
                                                    const float* __restrict__ beta,
                                                    const float* __restrict__ logb,
                                                    const float* __restrict__ logpi,
                                                    const float* __restrict__ PT,
                                                    const float* __restrict__ QT,
                                                    float* __restrict__ emis_part,
                                                    float* __restrict__ tran_part,
                                                    float* __restrict__ prior_part) {
    __shared__ float red[256];
    int t = blockIdx.x, j = threadIdx.x;
    float g = alpha[t * H + j] * beta[t * H + j];
    float G = block_reduce_sum(g, red);
    float gamma = g / G;
    float esum = block_reduce_sum(gamma * logb[t * H + j], red);
    if (t == 0) {
        float ps = block_reduce_sum(gamma * logpi[j], red);
        if (j == 0) {
            emis_part[0] = esum;
            tran_part[0] = 0.f;
            prior_part[0] = ps;
        }
    } else {
        float ap = alpha[(t - 1) * H + j];
        float Z = block_reduce_sum(ap * PT[t * H + j], red);
        float N = block_reduce_sum(ap * QT[t * H + j], red);
        if (j == 0) {
            emis_part[t] = esum;
            tran_part[t] = N / Z;
        }
    }
}

// ---------------- deterministic final reduction ----------------
__global__ __launch_bounds__(256) void final_reduce(const float* __restrict__ emis_part,
                                                    const float* __restrict__ tran_part,
                                                    const float* __restrict__ prior_part,
                                                    float* __restrict__ out) {
    __shared__ float red[256];
    int tid = threadIdx.x;
    float s = 0.f;
    for (int t = tid; t < TLEN; t += 256) s += emis_part[t] + tran_part[t];
    float tot = block_reduce_sum(s, red);
    if (tid == 0) out[0] = tot + prior_part[0];
}

// ---------------- host launcher ----------------
extern "C" void kernel_launch(void* const* d_in, const int* in_sizes, int n_in,
                              void* d_out, int out_size, void* d_ws, size_t ws_size,
                              hipStream_t stream) {
    (void)in_sizes; (void)n_in; (void)out_size; (void)ws_size;
    const float* obs = (const float*)d_in[0];   // [T,V]
    const float* sp  = (const float*)d_in[1];   // [H]
    const float* utr = (const float*)d_in[2];   // [H,H]
    const float* uem = (const float*)d_in[3];   // [H,V]

    float* p = (float*)d_ws;
    float* pi      = p; p += H;
    float* logpi   = p; p += H;
    float* trans   = p; p += H * H;
    float* transT  = p; p += H * H;
    float* tlogt   = p; p += H * H;
    float* emissP  = p; p += H * V;
    float* bprob   = p; p += TLEN * H;
    float* logb    = p; p += TLEN * H;
    float* alpha   = p; p += TLEN * H;
    float* beta    = p; p += TLEN * H;
    float* w       = p; p += TLEN * H;
    float* PT      = p; p += TLEN * H;
    float* QT      = p; p += TLEN * H;
    float* emis_part  = p; p += TLEN;
    float* tran_part  = p; p += TLEN;
    float* prior_part = p; p += 1;

    prior_softmax<<<1, 256, 0, stream>>>(sp, pi, logpi);
    trans_softmax<<<H, 256, 0, stream>>>(utr, trans, transT, tlogt);
    emiss_softmax<<<H, 256, 0, stream>>>(uem, emissP);
    logb_gemm<<<256, 256, 0, stream>>>(obs, emissP, logb, bprob);
    forward_scan<<<1, 256, 0, stream>>>(pi, trans, bprob, alpha);
    backward_scan<<<1, 256, 0, stream>>>(transT, bprob, beta, w);
    xi_gemm<<<256, 256, 0, stream>>>(trans, tlogt, w, PT, QT);
    per_t_reduce<<<TLEN, 256, 0, stream>>>(alpha, beta, logb, logpi, PT, QT,
                                           emis_part, tran_part, prior_part);
    final_reduce<<<1, 256, 0, stream>>>(emis_part, tran_part, prior_part, (float*)d_out);
}